// GaussianDeformer_54700703482238
// MI455X (gfx1250) — compile-verified
//
#include <hip/hip_runtime.h>
#include <hip/hip_bf16.h>

// ---------------------------------------------------------------------------
// GaussianDeformer on MI455X (gfx1250, wave32, WMMA).
// All dense math runs through V_WMMA_F32_16X16X4_F32 (exact fp32 path).
// Operand buffers are stored in WMMA-fragment-interleaved layout so every
// fragment is a single b64 load per lane.
// ---------------------------------------------------------------------------

#define BB 2
#define NN 4096
#define KNN_K 20

typedef float v2f __attribute__((ext_vector_type(2)));
typedef float v8f __attribute__((ext_vector_type(8)));

static __device__ __forceinline__ v8f wmma4(v2f a, v2f b, v8f c) {
  // D = A(16x4,f32) * B(4x16,f32) + C(16x16,f32)
  return __builtin_amdgcn_wmma_f32_16x16x4_f32(false, a, false, b, (short)0, c,
                                               false, false);
}

static __device__ __forceinline__ float leaky02(float y) {
  return y > 0.0f ? y : 0.2f * y;
}

// --------------------------- feature assembly ------------------------------
static __device__ __forceinline__ float gfeat_at(int n, int c,
                                                 const float* xyz,
                                                 const float* scale,
                                                 const float* rot,
                                                 const float* opacity,
                                                 const float* color) {
  if (c < 3)  return xyz[n * 3 + c];
  if (c < 6)  return scale[n * 3 + (c - 3)];
  if (c < 10) return rot[n * 4 + (c - 6)];
  if (c < 11) return opacity[n];
  return color[n * 3 + (c - 11)];
}

__global__ void build_feats_kernel(const float* __restrict__ xyz,
                                   const float* __restrict__ scale,
                                   const float* __restrict__ rot,
                                   const float* __restrict__ opacity,
                                   const float* __restrict__ color,
                                   const float* __restrict__ pose,
                                   const float* __restrict__ lbs,
                                   float* __restrict__ feats) {
  long t = (long)blockIdx.x * blockDim.x + threadIdx.x;
  if (t >= (long)BB * NN * 128) return;
  int c = (int)(t & 127);
  long bn = t >> 7;
  int n = (int)(bn % NN);
  int b = (int)(bn / NN);
  float v;
  if (c < 14)       v = gfeat_at(n, c, xyz, scale, rot, opacity, color);
  else if (c < 38)  v = lbs[n * 24 + (c - 14)];
  else if (c < 113) v = pose[b * 75 + (c - 38)];
  else              v = 0.0f;  // pad 113..127
  feats[t] = v;
}

// ---------------- transpose into WMMA B-fragment layout --------------------
// XTI[b][q=kc*2+hi][n][v] = X[b][n][4*kc + 2*hi + v]   (q in [0,CPAD/2))
// A lane's fragment pair (kb, kb+1) is contiguous -> one b64 load.
__global__ void transpose_kernel(const float* __restrict__ X, int ldX,
                                 float* __restrict__ XTI, int CPAD) {
  long t = (long)blockIdx.x * blockDim.x + threadIdx.x;
  long total = (long)BB * CPAD * NN;
  if (t >= total) return;
  long CB = (long)CPAD * NN;
  int b = (int)(t / CB);
  long r = t - (long)b * CB;
  int v = (int)(r & 1);
  long r2 = r >> 1;
  int n = (int)(r2 % NN);
  int q = (int)(r2 / NN);        // kc*2 + hi
  int hi = q & 1, kc = q >> 1;
  int c = 4 * kc + 2 * hi + v;
  XTI[t] = X[((long)b * NN + n) * ldX + c];
}

__global__ void sqnorm_kernel(const float* __restrict__ X, int ldX, int CPAD,
                              float* __restrict__ sq) {
  long t = (long)blockIdx.x * blockDim.x + threadIdx.x;
  if (t >= (long)BB * NN) return;
  const float* row = X + t * ldX;
  float s = 0.0f;
  for (int c = 0; c < CPAD; ++c) { float x = row[c]; s += x * x; }
  sq[t] = s;
}

// ------------------- weight repack (pads + interleave) ---------------------
// Logical Wpad[2*CPAD][D] stored as Wi[(c>>1)*2D + d*2 + (c&1)].
__global__ void repack_w_kernel(const float* __restrict__ W, int C, int CPAD,
                                int D, float* __restrict__ Wp) {
  long t = (long)blockIdx.x * blockDim.x + threadIdx.x;
  long total = (long)2 * CPAD * D;
  if (t >= total) return;
  int d = (int)(t % D);
  int c = (int)(t / D);
  float v = 0.0f;
  if (c < C)                          v = W[(long)c * D + d];
  else if (c >= CPAD && c < CPAD + C) v = W[(long)(C + c - CPAD) * D + d];
  Wp[(long)(c >> 1) * (2 * D) + (long)d * 2 + (c & 1)] = v;
}

// Dense [K][D] -> interleaved (no pads).
__global__ void repack_dense_kernel(const float* __restrict__ W, int K, int D,
                                    float* __restrict__ Wi) {
  long t = (long)blockIdx.x * blockDim.x + threadIdx.x;
  if (t >= (long)K * D) return;
  int d = (int)(t % D);
  int c = (int)(t / D);
  Wi[(long)(c >> 1) * (2 * D) + (long)d * 2 + (c & 1)] = W[t];
}

__global__ void pad_wo3_kernel(const float* __restrict__ W,
                               const float* __restrict__ bias,
                               float* __restrict__ Wp, float* __restrict__ bp) {
  int t = blockIdx.x * blockDim.x + threadIdx.x;
  if (t < 128 * 48) {
    int d = t % 48, k = t / 48;
    float v = (d < 38) ? W[k * 38 + d] : 0.0f;
    Wp[(k >> 1) * 96 + d * 2 + (k & 1)] = v;
  } else if (t < 128 * 48 + 48) {
    int d = t - 128 * 48;
    bp[d] = (d < 38) ? bias[d] : 0.0f;
  }
}

// ------------------------------- kNN (top-20) ------------------------------
// One wave per 16 query rows. Distances via WMMA on interleaved XTI
// (single-b64 fragments), per-lane top-20 lists in LDS with cached running
// min; the two lanes per row merged at the end.
template <int CPAD>
__global__ __launch_bounds__(256) void knn_kernel(
    const float* __restrict__ XTI, const float* __restrict__ sq,
    int* __restrict__ knn) {
  __shared__ float tile[8 * 256];
  __shared__ float lv[8 * 32 * KNN_K];
  __shared__ int   li[8 * 32 * KNN_K];

  int tid = threadIdx.x;
  int w = tid >> 5, lane = tid & 31;
  int lo = lane & 15, hi = lane >> 4;
  long r0 = (long)blockIdx.x * 128 + w * 16;  // global query row
  int b = (int)(r0 / NN);
  int nb = (int)(r0 % NN);
  const float* XTb = XTI + (long)b * CPAD * NN;

  // Cache the wave's A fragments (16 query rows) in registers (b64 loads).
  v2f avv[CPAD / 4];
#pragma unroll
  for (int kc = 0; kc < CPAD / 4; ++kc)
    avv[kc] = *(const v2f*)(XTb + ((long)(kc * 2 + hi) * NN + nb + lo) * 2);

  float sqr[8];
#pragma unroll
  for (int v = 0; v < 8; ++v) sqr[v] = sq[r0 + v + 8 * hi];

  float* mylv = &lv[(w * 32 + lane) * KNN_K];
  int* myli = &li[(w * 32 + lane) * KNN_K];
#pragma unroll
  for (int t = 0; t < KNN_K; ++t) { mylv[t] = -__builtin_inff(); myli[t] = 0; }
  float curmin = -__builtin_inff();
  int minpos = 0;

  for (int ct = 0; ct < NN / 16; ++ct) {
    int c0 = ct * 16;
    if (ct + 1 < NN / 16) {
      // gfx1250 global_prefetch for the next column tile
      __builtin_prefetch(
          (const void*)(XTb + ((long)hi * NN + c0 + 16 + lo) * 2), 0, 0);
    }
    v8f acc = {0.f, 0.f, 0.f, 0.f, 0.f, 0.f, 0.f, 0.f};
#pragma unroll
    for (int kc = 0; kc < CPAD / 4; ++kc) {
      v2f bf = *(const v2f*)(XTb + ((long)(kc * 2 + hi) * NN + c0 + lo) * 2);
      acc = wmma4(avv[kc], bf, acc);
    }
    float sqc = sq[(long)b * NN + c0 + lo];
#pragma unroll
    for (int v = 0; v < 8; ++v)
      tile[w * 256 + (v + 8 * hi) * 16 + lo] = 2.0f * acc[v] - sqr[v] - sqc;
    __syncthreads();
    // lane owns row (lane%16), columns [8*hi, 8*hi+8) -- read as b64 pairs
#pragma unroll
    for (int cc2 = 0; cc2 < 4; ++cc2) {
      v2f tv = *(const v2f*)&tile[w * 256 + lo * 16 + 8 * hi + cc2 * 2];
#pragma unroll
      for (int u = 0; u < 2; ++u) {
        float val = tv[u];
        int col = c0 + 8 * hi + cc2 * 2 + u;
        if (val > curmin) {
          mylv[minpos] = val; myli[minpos] = col;
          float mn = mylv[0]; int mp = 0;
#pragma unroll
          for (int t = 1; t < KNN_K; ++t) {
            float x = mylv[t];
            if (x < mn) { mn = x; mp = t; }
          }
          curmin = mn; minpos = mp;
        }
      }
    }
    __syncthreads();
  }

  // Merge the two per-row lane lists (lane, lane+16) -> top-20 of 40.
  __syncthreads();
  if (lane < 16) {
    for (int t = 0; t < KNN_K; ++t) {
      float best = -__builtin_inff();
      int bl = lane, bs = 0;
      for (int s = 0; s < 2 * KNN_K; ++s) {
        int ln = (s < KNN_K) ? lane : (lane + 16);
        int sl = (s < KNN_K) ? s : (s - KNN_K);
        float x = lv[(w * 32 + ln) * KNN_K + sl];
        if (x > best) { best = x; bl = ln; bs = sl; }
      }
      knn[(r0 + lane) * KNN_K + t] = li[(w * 32 + bl) * KNN_K + bs];
      lv[(w * 32 + bl) * KNN_K + bs] = -__builtin_inff();
    }
  }
}

// ------------------------------- EdgeConv ----------------------------------
// Block = 8 points = 160 edge rows = 10 WMMA row tiles. A frags gathered on
// the fly from [x_j - x_i | x_i] with b64 pair loads; h staged in LDS;
// BN + leaky + max-over-k reduced in-block (no global h tensor).
template <int CPAD, int D>
__global__ __launch_bounds__(256) void edge_conv_kernel(
    const float* __restrict__ X, int ldX, const int* __restrict__ knn,
    const float* __restrict__ Wp, const float* __restrict__ g,
    const float* __restrict__ bb, float* __restrict__ out, int ldo) {
  __shared__ float hbuf[160 * 64];
  int tid = threadIdx.x;
  int w = tid >> 5, lane = tid & 31;
  int lo = lane & 15, hi = lane >> 4;
  long P0 = (long)blockIdx.x * 8;
  const int K = 2 * CPAD;
  const float bnscale = rsqrtf(1.0f + 1e-5f);

  for (int dch = 0; dch < D / 64; ++dch) {
    for (int rt = w; rt < 10; rt += 8) {
      int ml = rt * 16 + lo;           // local edge row 0..159
      long gp = P0 + ml / 20;          // global point
      int kk = ml % 20;
      int b = (int)(gp / NN);
      long irow = gp;
      long jrow = (long)b * NN + knn[gp * KNN_K + kk];
      v8f acc[4];
#pragma unroll
      for (int nc = 0; nc < 4; ++nc)
        acc[nc] = (v8f){0.f, 0.f, 0.f, 0.f, 0.f, 0.f, 0.f, 0.f};

#pragma unroll 4
      for (int kc = 0; kc < K / 4; ++kc) {
        const bool first = (4 * kc) < CPAD;   // compile-time after unroll
        int kb = 4 * kc + 2 * hi;
        int cm = first ? kb : kb - CPAD;
        v2f xi = *(const v2f*)(X + irow * ldX + cm);
        v2f a;
        if (first) {
          v2f xj = *(const v2f*)(X + jrow * ldX + cm);
          a = xj - xi;
        } else {
          a = xi;
        }
#pragma unroll
        for (int nc = 0; nc < 4; ++nc) {
          int n0 = dch * 64 + nc * 16 + lo;
          v2f bf = *(const v2f*)(Wp + (long)(kb >> 1) * (2 * D) + n0 * 2);
          acc[nc] = wmma4(a, bf, acc[nc]);
        }
      }
#pragma unroll
      for (int nc = 0; nc < 4; ++nc)
#pragma unroll
        for (int v = 0; v < 8; ++v)
          hbuf[(rt * 16 + v + 8 * hi) * 64 + nc * 16 + lo] = acc[nc][v];
    }
    __syncthreads();
    // BN + leaky + max over k=20 for this 64-col chunk
    for (int t = tid; t < 8 * 64; t += 256) {
      int pl = t >> 6, dc = t & 63;
      int d = dch * 64 + dc;
      float gs = g[d] * bnscale;
      float bd = bb[d];
      float m = -__builtin_inff();
      for (int kk = 0; kk < KNN_K; ++kk) {
        float y = hbuf[(pl * 20 + kk) * 64 + dc] * gs + bd;
        m = fmaxf(m, leaky02(y));
      }
      out[(P0 + pl) * ldo + d] = m;
    }
    __syncthreads();
  }
}

// ----------------------------- generic GEMM --------------------------------
// One wave per (16-row x NC*16-col) output tile. W is interleaved.
// EPI: 0 = +bias, 1 = BN-scale + leaky (Wc5 path).
template <int NC, int EPI>
__global__ __launch_bounds__(32) void gemm_kernel(
    const float* __restrict__ A, int lda, const float* __restrict__ W, int ldw,
    const float* __restrict__ p1, const float* __restrict__ p2,
    float* __restrict__ out, int ldo, int K) {
  int lane = threadIdx.x;
  int lo = lane & 15, hi = lane >> 4;
  long r0 = (long)blockIdx.x * 16;
  int n0b = blockIdx.y * 16 * NC;
  v8f acc[NC];
#pragma unroll
  for (int nc = 0; nc < NC; ++nc)
    acc[nc] = (v8f){0.f, 0.f, 0.f, 0.f, 0.f, 0.f, 0.f, 0.f};

  for (int kc = 0; kc < K / 4; ++kc) {
    int kb = 4 * kc + 2 * hi;
    v2f a = *(const v2f*)(A + (r0 + lo) * lda + kb);
#pragma unroll
    for (int nc = 0; nc < NC; ++nc) {
      int n0 = n0b + nc * 16 + lo;
      v2f bf = *(const v2f*)(W + (long)(kb >> 1) * (2 * ldw) + n0 * 2);
      acc[nc] = wmma4(a, bf, acc[nc]);
    }
  }
  const float bnscale = rsqrtf(1.0f + 1e-5f);
#pragma unroll
  for (int nc = 0; nc < NC; ++nc) {
    int col = n0b + nc * 16 + lo;
#pragma unroll
    for (int v = 0; v < 8; ++v) {
      long row = r0 + v + 8 * hi;
      float y = acc[nc][v];
      if (EPI == 0) {
        y += p1[col];
      } else {
        y = leaky02(y * (p1[col] * bnscale) + p2[col]);
      }
      out[row * ldo + col] = y;
    }
  }
}

// -------------------------- layernorm + leaky ------------------------------
__global__ __launch_bounds__(32) void ln_leaky_kernel(
    float* __restrict__ h, const float* __restrict__ g,
    const float* __restrict__ b) {
  long row = blockIdx.x;
  int lane = threadIdx.x;
  float x[4], s = 0.f, ss = 0.f;
#pragma unroll
  for (int q = 0; q < 4; ++q) {
    x[q] = h[row * 128 + lane + 32 * q];
    s += x[q];
    ss += x[q] * x[q];
  }
#pragma unroll
  for (int off = 16; off >= 1; off >>= 1) {
    s += __shfl_xor(s, off, 32);
    ss += __shfl_xor(ss, off, 32);
  }
  float m = s * (1.0f / 128.0f);
  float var = ss * (1.0f / 128.0f) - m * m;
  float rs = rsqrtf(var + 1e-5f);
#pragma unroll
  for (int q = 0; q < 4; ++q) {
    int c = lane + 32 * q;
    float y = (x[q] - m) * rs * g[c] + b[c];
    h[row * 128 + c] = leaky02(y);
  }
}

// ------------------------------ final output -------------------------------
__global__ void assemble_kernel(const float* __restrict__ xyz,
                                const float* __restrict__ scale,
                                const float* __restrict__ rot,
                                const float* __restrict__ opacity,
                                const float* __restrict__ color,
                                const float* __restrict__ off,
                                float* __restrict__ out) {
  long t = (long)blockIdx.x * blockDim.x + threadIdx.x;
  if (t >= (long)BB * NN * 38) return;
  int c = (int)(t % 38);
  long bn = t / 38;
  int n = (int)(bn % NN);
  float base = (c < 14) ? gfeat_at(n, c, xyz, scale, rot, opacity, color) : 0.f;
  out[t] = base + off[bn * 48 + c];
}

// ---------------------------------------------------------------------------
// Host-side launch sequence
// ---------------------------------------------------------------------------
extern "C" void kernel_launch(void* const* d_in, const int* in_sizes, int n_in,
                              void* d_out, int out_size, void* d_ws,
                              size_t ws_size, hipStream_t stream) {
  (void)in_sizes; (void)n_in; (void)out_size; (void)ws_size;
  const float* xyz  = (const float*)d_in[0];
  const float* scl  = (const float*)d_in[1];
  const float* rot  = (const float*)d_in[2];
  const float* opa  = (const float*)d_in[3];
  const float* col  = (const float*)d_in[4];
  const float* pose = (const float*)d_in[5];
  const float* lbs  = (const float*)d_in[6];
  const float* Wc1 = (const float*)d_in[7],  *g1 = (const float*)d_in[8],  *b1 = (const float*)d_in[9];
  const float* Wc2 = (const float*)d_in[10], *g2 = (const float*)d_in[11], *b2 = (const float*)d_in[12];
  const float* Wc3 = (const float*)d_in[13], *g3 = (const float*)d_in[14], *b3 = (const float*)d_in[15];
  const float* Wc4 = (const float*)d_in[16], *g4 = (const float*)d_in[17], *b4 = (const float*)d_in[18];
  const float* Wc5 = (const float*)d_in[19], *g5 = (const float*)d_in[20], *b5 = (const float*)d_in[21];
  const float* Wo1 = (const float*)d_in[22], *bo1 = (const float*)d_in[23];
  const float* lg1 = (const float*)d_in[24], *lb1 = (const float*)d_in[25];
  const float* Wo2 = (const float*)d_in[26], *bo2 = (const float*)d_in[27];
  const float* lg2 = (const float*)d_in[28], *lb2 = (const float*)d_in[29];
  const float* Wo3 = (const float*)d_in[30], *bo3 = (const float*)d_in[31];
  float* out = (float*)d_out;

  // Workspace layout (floats)
  float* ws = (float*)d_ws;
  float* feats0 = ws;                              // [B*N][128]
  float* cat    = feats0 + (long)BB * NN * 128;    // [B*N][512]
  float* XT     = cat    + (long)BB * NN * 512;    // interleaved [B][CPAD<=128][N]
  float* sqb    = XT     + (long)BB * 128 * NN;    // [B*N]
  int*   knnI   = (int*)(sqb + (long)BB * NN);     // [B*N][20]
  float* Wpad   = (float*)(knnI + (long)BB * NN * KNN_K);  // interleaved [2*CPAD<=256][<=256]
  float* gf     = Wpad + 256 * 256;                // [B*N][128]
  float* h1     = gf   + (long)BB * NN * 128;
  float* h2     = h1   + (long)BB * NN * 128;
  float* off    = h2   + (long)BB * NN * 128;      // [B*N][48]
  float* W3p    = off  + (long)BB * NN * 48;       // interleaved [128][48]
  float* b3p    = W3p  + 128 * 48;                 // [48]
  float* Wc5i   = b3p  + 64;                       // interleaved [512][128]
  float* Wo1i   = Wc5i + 512 * 128;                // interleaved [128][128]
  float* Wo2i   = Wo1i + 128 * 128;                // interleaved [128][128]

  const int T = 256;
  long tot;

  // feats (113 logical, padded to 128)
  tot = (long)BB * NN * 128;
  build_feats_kernel<<<(tot + T - 1) / T, T, 0, stream>>>(
      xyz, scl, rot, opa, col, pose, lbs, feats0);

  // ---- EdgeConv layer 1: C=113 (pad 128), D=64 ----
  tot = (long)BB * 128 * NN;
  transpose_kernel<<<(tot + T - 1) / T, T, 0, stream>>>(feats0, 128, XT, 128);
  sqnorm_kernel<<<(BB * NN + T - 1) / T, T, 0, stream>>>(feats0, 128, 128, sqb);
  knn_kernel<128><<<BB * NN / 128, 256, 0, stream>>>(XT, sqb, knnI);
  tot = 2L * 128 * 64;
  repack_w_kernel<<<(tot + T - 1) / T, T, 0, stream>>>(Wc1, 113, 128, 64, Wpad);
  edge_conv_kernel<128, 64><<<BB * NN / 8, 256, 0, stream>>>(
      feats0, 128, knnI, Wpad, g1, b1, cat + 0, 512);

  // ---- EdgeConv layer 2: C=64, D=64 ----
  tot = (long)BB * 64 * NN;
  transpose_kernel<<<(tot + T - 1) / T, T, 0, stream>>>(cat + 0, 512, XT, 64);
  sqnorm_kernel<<<(BB * NN + T - 1) / T, T, 0, stream>>>(cat + 0, 512, 64, sqb);
  knn_kernel<64><<<BB * NN / 128, 256, 0, stream>>>(XT, sqb, knnI);
  tot = 2L * 64 * 64;
  repack_w_kernel<<<(tot + T - 1) / T, T, 0, stream>>>(Wc2, 64, 64, 64, Wpad);
  edge_conv_kernel<64, 64><<<BB * NN / 8, 256, 0, stream>>>(
      cat + 0, 512, knnI, Wpad, g2, b2, cat + 64, 512);

  // ---- EdgeConv layer 3: C=64, D=128 ----
  tot = (long)BB * 64 * NN;
  transpose_kernel<<<(tot + T - 1) / T, T, 0, stream>>>(cat + 64, 512, XT, 64);
  sqnorm_kernel<<<(BB * NN + T - 1) / T, T, 0, stream>>>(cat + 64, 512, 64, sqb);
  knn_kernel<64><<<BB * NN / 128, 256, 0, stream>>>(XT, sqb, knnI);
  tot = 2L * 64 * 128;
  repack_w_kernel<<<(tot + T - 1) / T, T, 0, stream>>>(Wc3, 64, 64, 128, Wpad);
  edge_conv_kernel<64, 128><<<BB * NN / 8, 256, 0, stream>>>(
      cat + 64, 512, knnI, Wpad, g3, b3, cat + 128, 512);

  // ---- EdgeConv layer 4: C=128, D=256 ----
  tot = (long)BB * 128 * NN;
  transpose_kernel<<<(tot + T - 1) / T, T, 0, stream>>>(cat + 128, 512, XT, 128);
  sqnorm_kernel<<<(BB * NN + T - 1) / T, T, 0, stream>>>(cat + 128, 512, 128, sqb);
  knn_kernel<128><<<BB * NN / 128, 256, 0, stream>>>(XT, sqb, knnI);
  tot = 2L * 128 * 256;
  repack_w_kernel<<<(tot + T - 1) / T, T, 0, stream>>>(Wc4, 128, 128, 256, Wpad);
  edge_conv_kernel<128, 256><<<BB * NN / 8, 256, 0, stream>>>(
      cat + 128, 512, knnI, Wpad, g4, b4, cat + 256, 512);

  // ---- interleave head weights ----
  tot = 512L * 128;
  repack_dense_kernel<<<(tot + T - 1) / T, T, 0, stream>>>(Wc5, 512, 128, Wc5i);
  tot = 128L * 128;
  repack_dense_kernel<<<(tot + T - 1) / T, T, 0, stream>>>(Wo1, 128, 128, Wo1i);
  repack_dense_kernel<<<(tot + T - 1) / T, T, 0, stream>>>(Wo2, 128, 128, Wo2i);
  pad_wo3_kernel<<<(128 * 48 + 48 + T - 1) / T, T, 0, stream>>>(Wo3, bo3, W3p, b3p);

  // ---- gf = leaky(BN(cat @ Wc5)) : K=512 -> D=128 ----
  gemm_kernel<4, 1><<<dim3(BB * NN / 16, 2), 32, 0, stream>>>(
      cat, 512, Wc5i, 128, g5, b5, gf, 128, 512);

  // ---- head MLP ----
  gemm_kernel<4, 0><<<dim3(BB * NN / 16, 2), 32, 0, stream>>>(
      gf, 128, Wo1i, 128, bo1, nullptr, h1, 128, 128);
  ln_leaky_kernel<<<BB * NN, 32, 0, stream>>>(h1, lg1, lb1);

  gemm_kernel<4, 0><<<dim3(BB * NN / 16, 2), 32, 0, stream>>>(
      h1, 128, Wo2i, 128, bo2, nullptr, h2, 128, 128);
  ln_leaky_kernel<<<BB * NN, 32, 0, stream>>>(h2, lg2, lb2);

  gemm_kernel<3, 0><<<dim3(BB * NN / 16, 1), 32, 0, stream>>>(
      h2, 128, W3p, 48, b3p, nullptr, off, 48, 128);

  tot = (long)BB * NN * 38;
  assemble_kernel<<<(tot + T - 1) / T, T, 0, stream>>>(
      xyz, scl, rot, opa, col, off, out);
}